// Local_Fai_score_12910671692201
// MI455X (gfx1250) — compile-verified
//
#include <hip/hip_runtime.h>
#include <hip/hip_bf16.h>
#include <math.h>

typedef float v2f __attribute__((ext_vector_type(2)));
typedef float v8f __attribute__((ext_vector_type(8)));

#define WAVE 32

// ---------------------------------------------------------------------------
// Batched f32 WMMA GEMM:  C[b1,b2] = alpha * A[b1,b2] * B[b1,b2]
//
// KW == 0 : plain GEMM. A: MxK row-major (lda). B: NN = KxN (ldb) or
//           NT = NxK (ldb) when TB. C: MxN (ldc).
// KW  > 0 : convolution-as-GEMM over pre-gathered embeddings.
//           A = E [.., D] row-major (lda = D), logical GEMM row r = m*P + p
//           reads E rows m*L + p + kk for kk in [0, KW).
//           B = W [N, D*KW] row-major (ldb = D*KW), tap kk at column d*KW+kk.
//           K parameter = D. Always NT semantics.
//
// One wave computes one 16x16 C tile with V_WMMA_F32_16X16X4_F32.
// Out-of-range rows/cols: load addresses are clamped (garbage values are
// computed but only land in unstored C elements) -> branchless hot loop.
// Two accumulators break the WMMA C-chain dependency.
// ---------------------------------------------------------------------------
template<int KW, bool TB>
__global__ __launch_bounds__(WAVE) void wmma_gemm(
    const float* __restrict__ A, const float* __restrict__ Bm, float* __restrict__ C,
    int M, int N, int K, int lda, int ldb, int ldc, float alpha,
    long sA1, long sB1, long sC1, int nB2, long sA2, long sB2, long sC2,
    int P, int L)
{
    const int lane = threadIdx.x;
    const int lo = lane & 15;
    const int hi = lane >> 4;
    const int kofs = 2 * hi;               // this lane supplies K = k0+kofs, +1
    const int b  = blockIdx.z;
    const int b1 = b / nB2;
    const int b2 = b - b1 * nB2;

    const float* Ab = A  + b1 * sA1 + b2 * sA2;
    const float* Bb = Bm + b1 * sB1 + b2 * sB2;
    float*       Cb = C  + b1 * sC1 + b2 * sC2;

    const int row  = blockIdx.y * 16 + lo;
    const int col  = blockIdx.x * 16 + lo;
    const int rowC = row < M ? row : M - 1;   // clamped (address-safe)
    const int colC = col < N ? col : N - 1;

    v8f acc0 = {}, acc1 = {};

    if (KW == 0) {
        const float* ar = Ab + (long)rowC * lda;
        const int nk = K >> 2;
        int kt = 0;
        for (; kt + 2 <= nk; kt += 2) {
            const int ka = (kt << 2) + kofs;
            const int kb = ka + 4;
            v2f a0, a1, b0, b1v;
            a0.x = ar[ka];     a0.y = ar[ka + 1];
            a1.x = ar[kb];     a1.y = ar[kb + 1];
            if (TB) {
                const float* br = Bb + (long)colC * ldb;
                b0.x  = br[ka]; b0.y  = br[ka + 1];
                b1v.x = br[kb]; b1v.y = br[kb + 1];
            } else {
                b0.x  = Bb[(long)ka * ldb + colC];
                b0.y  = Bb[(long)(ka + 1) * ldb + colC];
                b1v.x = Bb[(long)kb * ldb + colC];
                b1v.y = Bb[(long)(kb + 1) * ldb + colC];
            }
            acc0 = __builtin_amdgcn_wmma_f32_16x16x4_f32(false, a0, false, b0,  (short)0, acc0, false, false);
            acc1 = __builtin_amdgcn_wmma_f32_16x16x4_f32(false, a1, false, b1v, (short)0, acc1, false, false);
        }
        if (kt < nk) {
            const int ka = (kt << 2) + kofs;
            v2f a0, b0;
            a0.x = ar[ka]; a0.y = ar[ka + 1];
            if (TB) {
                const float* br = Bb + (long)colC * ldb;
                b0.x = br[ka]; b0.y = br[ka + 1];
            } else {
                b0.x = Bb[(long)ka * ldb + colC];
                b0.y = Bb[(long)(ka + 1) * ldb + colC];
            }
            acc0 = __builtin_amdgcn_wmma_f32_16x16x4_f32(false, a0, false, b0, (short)0, acc0, false, false);
        }
    } else {
        // conv-as-GEMM: sum over taps kk; inner loop over d (K = D)
        const int mm = rowC / P;
        const int pp = rowC - mm * P;
        const float* aBase = Ab + ((long)mm * L + pp) * lda;   // E row, kk = 0
        const float* bBase = Bb + (long)colC * ldb;            // W row for filter
        const int nk = K >> 2;
        #pragma unroll
        for (int kk = 0; kk < KW; ++kk) {
            const float* ar = aBase + (long)kk * lda;
            const float* br = bBase + kk;
            int kt = 0;
            for (; kt + 2 <= nk; kt += 2) {
                const int da = (kt << 2) + kofs;
                const int db = da + 4;
                v2f a0, a1, b0, b1v;
                a0.x = ar[da]; a0.y = ar[da + 1];
                a1.x = ar[db]; a1.y = ar[db + 1];
                b0.x  = br[(long)da * KW];       b0.y  = br[(long)(da + 1) * KW];
                b1v.x = br[(long)db * KW];       b1v.y = br[(long)(db + 1) * KW];
                acc0 = __builtin_amdgcn_wmma_f32_16x16x4_f32(false, a0, false, b0,  (short)0, acc0, false, false);
                acc1 = __builtin_amdgcn_wmma_f32_16x16x4_f32(false, a1, false, b1v, (short)0, acc1, false, false);
            }
            if (kt < nk) {
                const int da = (kt << 2) + kofs;
                v2f a0, b0;
                a0.x = ar[da]; a0.y = ar[da + 1];
                b0.x = br[(long)da * KW]; b0.y = br[(long)(da + 1) * KW];
                acc0 = __builtin_amdgcn_wmma_f32_16x16x4_f32(false, a0, false, b0, (short)0, acc0, false, false);
            }
        }
    }

    v8f acc = acc0 + acc1;
    if (col < N) {
        #pragma unroll
        for (int r = 0; r < 8; ++r) {
            int cm = blockIdx.y * 16 + r + 8 * hi;
            if (cm < M) Cb[(long)cm * ldc + col] = alpha * acc[r];
        }
    }
}

template<int KW, bool TB>
static inline void launch_gemm(hipStream_t s,
    const float* A, const float* B, float* C,
    int M, int N, int K, int lda, int ldb, int ldc, float alpha,
    int nB1, long sA1, long sB1, long sC1,
    int nB2, long sA2, long sB2, long sC2,
    int P = 1, int L = 1)
{
    dim3 grid((N + 15) / 16, (M + 15) / 16, nB1 * nB2);
    wmma_gemm<KW, TB><<<grid, dim3(WAVE, 1, 1), 0, s>>>(
        A, B, C, M, N, K, lda, ldb, ldc, alpha,
        sA1, sB1, sC1, nB2, sA2, sB2, sC2, P, L);
}

// ---------------------------------------------------------------------------
// Embedding row gather: out[i, :] = table[ids[i], :]
// ---------------------------------------------------------------------------
__global__ void gather_rows(const int* __restrict__ ids, const float* __restrict__ table,
                            float* __restrict__ out, int nrows, int D)
{
    long i = (long)blockIdx.x * blockDim.x + threadIdx.x;
    long total = (long)nrows * D;
    if (i >= total) return;
    long r = i / D;
    int  c = (int)(i - r * D);
    out[i] = table[(long)ids[r] * D + c];
}

// feat[m,f] = mean_p relu(Y[m*P+p, f] + bias[f])
__global__ void relu_bias_avg(const float* __restrict__ Y, const float* __restrict__ bias,
                              float* __restrict__ feat, int Mq, int P, int F)
{
    int i = blockIdx.x * blockDim.x + threadIdx.x;
    if (i >= Mq * F) return;
    int m = i / F, f = i - m * F;
    float bb = bias[f];
    float s = 0.f;
    for (int p = 0; p < P; ++p) {
        float v = Y[((long)(m * P + p)) * F + f] + bb;
        s += v > 0.f ? v : 0.f;
    }
    feat[i] = s / (float)P;
}

// In-place row softmax, one wave per row.
__global__ __launch_bounds__(WAVE) void softmax_rows(float* __restrict__ X, int rows, int L)
{
    int r = blockIdx.x;
    if (r >= rows) return;
    float* x = X + (long)r * L;
    int lane = threadIdx.x;
    float mx = -3.4e38f;
    for (int i = lane; i < L; i += WAVE) mx = fmaxf(mx, x[i]);
    for (int o = 16; o; o >>= 1) mx = fmaxf(mx, __shfl_xor(mx, o, WAVE));
    float s = 0.f;
    for (int i = lane; i < L; i += WAVE) { float e = expf(x[i] - mx); x[i] = e; s += e; }
    for (int o = 16; o; o >>= 1) s += __shfl_xor(s, o, WAVE);
    float inv = 1.f / s;
    for (int i = lane; i < L; i += WAVE) x[i] *= inv;
}

// out[n, d] = max_q X[n, q, d]
__global__ void maxpool_axis(const float* __restrict__ X, float* __restrict__ out,
                             int nB, int Q, int D)
{
    int i = blockIdx.x * blockDim.x + threadIdx.x;
    if (i >= nB * D) return;
    int n = i / D, d = i - n * D;
    const float* p = X + (long)n * Q * D + d;
    float m = p[0];
    for (int q = 1; q < Q; ++q) m = fmaxf(m, p[(long)q * D]);
    out[i] = m;
}

// hidden[n,j] = relu(b1[j] + sum_c comb[n,c] * w1[c,j]),  comb = [attdb | attbd]
__global__ void att_hidden_k(const float* __restrict__ attdb, const float* __restrict__ attbd,
                             const float* __restrict__ w1, const float* __restrict__ b1,
                             float* __restrict__ hidden)
{
    int i = blockIdx.x * blockDim.x + threadIdx.x;
    if (i >= 16 * 300) return;
    int n = i / 300, j = i - n * 300;
    float s = b1[j];
    const float* a = attdb + n * 300;
    const float* b = attbd + n * 300;
    for (int c = 0; c < 300; ++c) s += a[c] * w1[c * 300 + j];
    for (int c = 0; c < 300; ++c) s += b[c] * w1[(300 + c) * 300 + j];
    hidden[i] = s > 0.f ? s : 0.f;
}

// att_score[n] = sigmoid(sum_j hidden[n,j]*w2[j] + b2)
__global__ __launch_bounds__(WAVE) void att_out_k(const float* __restrict__ hidden,
                                                  const float* __restrict__ w2,
                                                  const float* __restrict__ b2,
                                                  float* __restrict__ out)
{
    int n = blockIdx.x, lane = threadIdx.x;
    float s = 0.f;
    for (int j = lane; j < 300; j += WAVE) s += hidden[n * 300 + j] * w2[j];
    for (int o = 16; o; o >>= 1) s += __shfl_xor(s, o, WAVE);
    if (lane == 0) out[n] = 1.f / (1.f + expf(-(s + b2[0])));
}

// tokmax[m, t] = max_n tok[m, n, t]
__global__ void tok_max_n(const float* __restrict__ tok, float* __restrict__ out)
{
    int i = blockIdx.x * blockDim.x + threadIdx.x;
    if (i >= 64 * 50) return;
    int m = i / 50, t = i - m * 50;
    const float* p = tok + (long)m * 800 + t;
    float v = p[0];
    for (int n = 1; n < 16; ++n) v = fmaxf(v, p[n * 50]);
    out[i] = v;
}

// ctxv[m, d] = sum_t p[m,t] * ctx_e[m, t, d]
__global__ void ctx_weighted(const float* __restrict__ p, const float* __restrict__ ctx_e,
                             float* __restrict__ ctxv)
{
    int i = blockIdx.x * blockDim.x + threadIdx.x;
    if (i >= 64 * 300) return;
    int m = i / 300, d = i - m * 300;
    const float* pe = ctx_e + (long)m * 15000 + d;
    const float* pw = p + m * 50;
    float s = 0.f;
    for (int t = 0; t < 50; ++t) s += pw[t] * pe[(long)t * 300];
    ctxv[i] = s;
}

// local_att[m,n] = dot(cand_e[m,n,:], ctxv[m,:])  (one wave per (m,n))
__global__ __launch_bounds__(WAVE) void local_att_k(const float* __restrict__ cand_e,
                                                    const float* __restrict__ ctxv,
                                                    float* __restrict__ out)
{
    int idx = blockIdx.x;
    int m = idx >> 4;
    int lane = threadIdx.x;
    const float* a = cand_e + (long)idx * 300;
    const float* b = ctxv + (long)m * 300;
    float s = 0.f;
    for (int d = lane; d < 300; d += WAVE) s += a[d] * b[d];
    for (int o = 16; o; o >>= 1) s += __shfl_xor(s, o, WAVE);
    if (lane == 0) out[idx] = s;
}

// cos_st / cos_ct, one wave per (m,n)
__global__ __launch_bounds__(WAVE) void cos_mn_k(const float* __restrict__ ms,
                                                 const float* __restrict__ mc,
                                                 const float* __restrict__ title,
                                                 float* __restrict__ cosst,
                                                 float* __restrict__ cosct)
{
    int idx = blockIdx.x;
    int m = idx >> 4, n = idx & 15;
    int lane = threadIdx.x;
    const float* a = ms + (long)m * 768;
    const float* c = mc + (long)m * 768;
    const float* t = title + (long)n * 768;
    float dms = 0.f, dmc = 0.f, nm = 0.f, nc = 0.f, nt = 0.f;
    for (int i = lane; i < 768; i += WAVE) {
        float av = a[i], cv = c[i], tv = t[i];
        dms += av * tv; dmc += cv * tv;
        nm += av * av; nc += cv * cv; nt += tv * tv;
    }
    for (int o = 16; o; o >>= 1) {
        dms += __shfl_xor(dms, o, WAVE); dmc += __shfl_xor(dmc, o, WAVE);
        nm  += __shfl_xor(nm,  o, WAVE); nc  += __shfl_xor(nc,  o, WAVE);
        nt  += __shfl_xor(nt,  o, WAVE);
    }
    if (lane == 0) {
        float tn = fmaxf(sqrtf(nt), 1e-6f);
        cosst[idx] = dms / (fmaxf(sqrtf(nm), 1e-6f) * tn);
        cosct[idx] = dmc / (fmaxf(sqrtf(nc), 1e-6f) * tn);
    }
}

__global__ __launch_bounds__(WAVE) void cos_dt_k(const float* __restrict__ bert,
                                                 const float* __restrict__ title,
                                                 float* __restrict__ out)
{
    int n = blockIdx.x, lane = threadIdx.x;
    const float* t = title + (long)n * 768;
    float dd = 0.f, nb = 0.f, nt = 0.f;
    for (int i = lane; i < 768; i += WAVE) {
        float b = bert[i], tv = t[i];
        dd += b * tv; nb += b * b; nt += tv * tv;
    }
    for (int o = 16; o; o >>= 1) {
        dd += __shfl_xor(dd, o, WAVE); nb += __shfl_xor(nb, o, WAVE); nt += __shfl_xor(nt, o, WAVE);
    }
    if (lane == 0) out[n] = dd / (fmaxf(sqrtf(nb), 1e-6f) * fmaxf(sqrtf(nt), 1e-6f));
}

// Final: score -> standardize (ddof=1) -> softmax -> uniform renorm. One wave / mention.
__global__ __launch_bounds__(WAVE) void finalize_k(
    const float* __restrict__ hand, const float* __restrict__ prior,
    const float* __restrict__ latt, const float* __restrict__ attsc,
    const float* __restrict__ cosst, const float* __restrict__ cosdt,
    const float* __restrict__ cosct, const float* __restrict__ w,
    const float* __restrict__ bptr, float* __restrict__ out)
{
    int m = blockIdx.x;
    int lane = threadIdx.x;
    int n = lane & 15;                 // lanes 16..31 mirror lanes 0..15
    long i = (long)m * 16 + n;
    float score = w[0] * hand[n] + w[1] * prior[n] + w[2] * latt[i] + w[3] * attsc[n]
                + w[4] * cosst[i] + w[5] * cosdt[n] + w[6] * cosct[i] + bptr[0];

    float tot = score;
    for (int o = 8; o; o >>= 1) tot += __shfl_xor(tot, o, 16);
    float mu = tot * (1.f / 16.f);
    float d = score - mu;
    float ssq = d * d;
    for (int o = 8; o; o >>= 1) ssq += __shfl_xor(ssq, o, 16);
    float sd = sqrtf(ssq / 15.f);      // unbiased (ddof=1)
    float z = d / sd;

    float zmx = z;
    for (int o = 8; o; o >>= 1) zmx = fmaxf(zmx, __shfl_xor(zmx, o, 16));
    float e = expf(z - zmx);
    float se = e;
    for (int o = 8; o; o >>= 1) se += __shfl_xor(se, o, 16);
    float sm = e / se;

    float zmn = z;
    for (int o = 8; o; o >>= 1) zmn = fminf(zmn, __shfl_xor(zmn, o, 16));
    float u = (z + 1.f - zmn) / (zmx - zmn);
    float su = u;
    for (int o = 8; o; o >>= 1) su += __shfl_xor(su, o, 16);
    u /= su;

    if (lane < 16) {
        out[i]        = z;
        out[1024 + i] = sm;
        out[2048 + i] = u;
    }
}

// ---------------------------------------------------------------------------
extern "C" void kernel_launch(void* const* d_in, const int* in_sizes, int n_in,
                              void* d_out, int out_size, void* d_ws, size_t ws_size,
                              hipStream_t stream)
{
    (void)in_sizes; (void)n_in; (void)out_size; (void)ws_size;

    const int*   mention_vec  = (const int*)  d_in[3];   // [64,4]
    const int*   context_vec  = (const int*)  d_in[4];   // [64,50]
    const int*   doc_vec      = (const int*)  d_in[5];   // [100]
    const float* title_vec    = (const float*)d_in[6];   // [16,768]
    const int*   body_vec     = (const int*)  d_in[7];   // [16,200]
    const float* m2c_prior    = (const float*)d_in[9];   // [16]
    const int*   men2cands    = (const int*)  d_in[10];  // [64,16]
    const int*   contexts_ids = (const int*)  d_in[11];  // [64,50]
    const float* hand_feat    = (const float*)d_in[12];  // [16,1]
    const float* bert_doc     = (const float*)d_in[13];  // [1,768]
    const float* embed        = (const float*)d_in[14];  // [100000,300]
    const float* conv_ms_w    = (const float*)d_in[15];  // [768,300,2]
    const float* conv_ms_b    = (const float*)d_in[16];  // [768]
    const float* conv_mc_w    = (const float*)d_in[17];  // [768,300,5]
    const float* conv_mc_b    = (const float*)d_in[18];  // [768]
    const float* lw           = (const float*)d_in[19];  // [7]
    const float* lb           = (const float*)d_in[20];  // scalar
    const float* att_w1       = (const float*)d_in[21];  // [600,300]
    const float* att_b1       = (const float*)d_in[22];  // [300]
    const float* att_w2       = (const float*)d_in[23];  // [300,1]
    const float* att_b2       = (const float*)d_in[24];  // scalar
    float* out = (float*)d_out;                          // 3 x [64,16] (z, sm, u)

    // ---- workspace carve-up (~34 MB) ----
    char* ws = (char*)d_ws;
    size_t off = 0;
    auto take = [&](size_t nfloats) -> float* {
        float* p = (float*)(ws + off);
        off = (off + nfloats * sizeof(float) + 255) & ~(size_t)255;
        return p;
    };
    float* Yms    = take((size_t)192 * 768);      // conv ms pre-pool
    float* Ymc    = take((size_t)2944 * 768);     // conv mc pre-pool
    float* msf    = take((size_t)64 * 768);
    float* mcf    = take((size_t)64 * 768);
    float* mvec_e = take((size_t)256 * 300);      // mention_vec embeddings
    float* cvec_e = take((size_t)3200 * 300);     // context_vec embeddings
    float* doc_e  = take((size_t)100 * 300);
    float* body_e = take((size_t)3200 * 300);
    float* ctx_e  = take((size_t)3200 * 300);
    float* cand_e = take((size_t)1024 * 300);
    float* logits = take((size_t)16 * 5 * 100 * 200);  // reused both directions
    float* attout = take((size_t)16 * 200 * 300);      // reused both directions
    float* attdb  = take((size_t)16 * 300);
    float* attbd  = take((size_t)16 * 300);
    float* hidden = take((size_t)16 * 300);
    float* attsc  = take((size_t)16);
    float* tokb   = take((size_t)64 * 16 * 50);
    float* tokmx  = take((size_t)64 * 50);
    float* ctxv   = take((size_t)64 * 300);
    float* latt   = take((size_t)64 * 16);
    float* cosst  = take((size_t)64 * 16);
    float* cosct  = take((size_t)64 * 16);
    float* cosdt  = take((size_t)16);

    const float attn_scale = 0.12909944487358056f;  // 1/sqrt(60)

    // ---- embedding gathers ----
    {
        int t;
        t = 256 * 300;  gather_rows<<<(t + 255) / 256, 256, 0, stream>>>(mention_vec,  embed, mvec_e, 256,  300);
        t = 3200 * 300; gather_rows<<<(t + 255) / 256, 256, 0, stream>>>(context_vec,  embed, cvec_e, 3200, 300);
        t = 100 * 300;  gather_rows<<<(t + 255) / 256, 256, 0, stream>>>(doc_vec,      embed, doc_e,  100,  300);
        t = 3200 * 300; gather_rows<<<(t + 255) / 256, 256, 0, stream>>>(body_vec,     embed, body_e, 3200, 300);
        t = 3200 * 300; gather_rows<<<(t + 255) / 256, 256, 0, stream>>>(contexts_ids, embed, ctx_e,  3200, 300);
        t = 1024 * 300; gather_rows<<<(t + 255) / 256, 256, 0, stream>>>(men2cands,    embed, cand_e, 1024, 300);
    }

    // ---- conv-as-GEMM, tap-sum formulation (WMMA f32, division-free) ----
    // ms: rows 64*3, D=300, KW=2; W row stride 600
    launch_gemm<2, true>(stream, mvec_e, conv_ms_w, Yms,
                         192, 768, 300, 300, 600, 768, 1.f,
                         1, 0, 0, 0, 1, 0, 0, 0, /*P=*/3, /*L=*/4);
    // mc: rows 64*46, D=300, KW=5; W row stride 1500
    launch_gemm<5, true>(stream, cvec_e, conv_mc_w, Ymc,
                         2944, 768, 300, 300, 1500, 768, 1.f,
                         1, 0, 0, 0, 1, 0, 0, 0, /*P=*/46, /*L=*/50);
    relu_bias_avg<<<(64 * 768 + 255) / 256, 256, 0, stream>>>(Yms, conv_ms_b, msf, 64, 3,  768);
    relu_bias_avg<<<(64 * 768 + 255) / 256, 256, 0, stream>>>(Ymc, conv_mc_b, mcf, 64, 46, 768);

    // ---- attention direction 1: q = doc[100], k = v = body[n][200] ----
    launch_gemm<0, true>(stream, doc_e, body_e, logits,
                         100, 200, 60, 300, 300, 200, attn_scale,
                         /*B1=n*/16, 0,      60000, 100000,
                         /*B2=h*/5,  60,     60,    20000);
    softmax_rows<<<16 * 5 * 100, WAVE, 0, stream>>>(logits, 16 * 5 * 100, 200);
    launch_gemm<0, false>(stream, logits, body_e, attout,
                          100, 60, 200, 200, 300, 300, 1.f,
                          16, 100000, 60000, 30000,
                          5,  20000,  60,    60);
    maxpool_axis<<<(16 * 300 + 255) / 256, 256, 0, stream>>>(attout, attdb, 16, 100, 300);

    // ---- attention direction 2: q = body[n][200], k = v = doc[100] ----
    launch_gemm<0, true>(stream, body_e, doc_e, logits,
                         200, 100, 60, 300, 300, 100, attn_scale,
                         16, 60000, 0,  100000,
                         5,  60,    60, 20000);
    softmax_rows<<<16 * 5 * 200, WAVE, 0, stream>>>(logits, 16 * 5 * 200, 100);
    launch_gemm<0, false>(stream, logits, doc_e, attout,
                          200, 60, 100, 100, 300, 300, 1.f,
                          16, 100000, 0,  60000,
                          5,  20000,  60, 60);
    maxpool_axis<<<(16 * 300 + 255) / 256, 256, 0, stream>>>(attout, attbd, 16, 200, 300);

    // ---- attention MLP head ----
    att_hidden_k<<<(16 * 300 + 255) / 256, 256, 0, stream>>>(attdb, attbd, att_w1, att_b1, hidden);
    att_out_k<<<16, WAVE, 0, stream>>>(hidden, att_w2, att_b2, attsc);

    // ---- LocalCtxAttRanker ----
    launch_gemm<0, true>(stream, cand_e, ctx_e, tokb,
                         16, 50, 300, 300, 300, 50, 1.f,
                         64, 4800, 15000, 800,
                         1, 0, 0, 0);
    tok_max_n<<<(64 * 50 + 255) / 256, 256, 0, stream>>>(tokb, tokmx);
    softmax_rows<<<64, WAVE, 0, stream>>>(tokmx, 64, 50);
    ctx_weighted<<<(64 * 300 + 255) / 256, 256, 0, stream>>>(tokmx, ctx_e, ctxv);
    local_att_k<<<1024, WAVE, 0, stream>>>(cand_e, ctxv, latt);

    // ---- cosine features ----
    cos_mn_k<<<1024, WAVE, 0, stream>>>(msf, mcf, title_vec, cosst, cosct);
    cos_dt_k<<<16, WAVE, 0, stream>>>(bert_doc, title_vec, cosdt);

    // ---- final score / standardize / softmax / renorm ----
    finalize_k<<<64, WAVE, 0, stream>>>(hand_feat, m2c_prior, latt, attsc,
                                        cosst, cosdt, cosct, lw, lb, out);
}